// RefinedQuantumEntanglementLayer_64209761075287
// MI455X (gfx1250) — compile-verified
//
#include <hip/hip_runtime.h>

// RefinedQuantumEntanglementLayer for MI455X (gfx1250):
//   U = expm(M_r - M_r^T) @ expm(M_i - M_i^T)   (scaling-and-squaring + Taylor)
//   out = (l2norm(x) @ U + bias)^2 * softmax(gate)
// All GEMMs run on V_WMMA_F32_16X16X4_F32 (fp32 accuracy). B panels are staged
// into LDS *transposed* via async global->LDS DMA (ASYNCcnt), double-buffered
// so the DMA overlaps the WMMA pipe; waves consume B fragments as ds_load_b64.

#define DQ 1024
#define DDQ ((size_t)DQ * (size_t)DQ)
#define NROWS (8 * 4096) /* 32768 */
#define TAYLOR_ORDER 8
#define SQUARINGS 6
#define EXPM_SCALE (1.0f / 64.0f) /* 2^-SQUARINGS */

#define KC 32        /* K-depth of one LDS B tile  */
#define BN 32        /* N-width of one block tile  */
#define BM 128       /* M-height of one block tile */
#define NTHREADS 128 /* 4 waves */

typedef __attribute__((ext_vector_type(2))) float v2f;
typedef __attribute__((ext_vector_type(8))) float v8f;

__device__ __forceinline__ v8f wmma_f32_4(v2f a, v2f b, v8f c) {
  // 8 args: (neg_a, A, neg_b, B, c_mod, C, reuse_a, reuse_b)
  return __builtin_amdgcn_wmma_f32_16x16x4_f32(false, a, false, b, (short)0, c,
                                               false, false);
}

// Async DMA: copies 4 bytes from the global address gaddr to the LDS address lds_off, tracked by ASYNCcnt.
__device__ __forceinline__ void async_b32_to_lds(unsigned lds_off,
                                                 const float* gaddr) {
  asm volatile("global_load_async_to_lds_b32 %0, %1, off" ::"v"(lds_off),
               "v"((unsigned long long)(uintptr_t)gaddr)
               : "memory");
}

__device__ __forceinline__ void wait_async0() {
  asm volatile("s_wait_asynccnt 0x0" ::: "memory");
}

// Issue the async copy of one KCxBN B tile, transposed into LDS as Bt[n][k].
// Global side: 32 consecutive lanes read 32 consecutive columns (coalesced).
__device__ __forceinline__ void issue_btile(const float* __restrict__ B, int N,
                                            int k0, int n0, float* BtBuf,
                                            int tid) {
#pragma unroll
  for (int i = 0; i < (KC * BN) / NTHREADS; ++i) {
    const int e = tid + i * NTHREADS;
    const int k = e >> 5;      // row within tile
    const int n = e & (BN - 1);
    const float* gp = B + (size_t)(k0 + k) * N + (n0 + n);
    const unsigned lds = (unsigned)(uintptr_t)(BtBuf + n * KC + k);
    async_b32_to_lds(lds, gp);
  }
}

// Core: one wave computes a 64x16 sub-tile (4 f32-WMMA accumulators).
// Block = 4 waves in a 2x2 grid over a 128x32 tile; B via double-buffered LDS.
__device__ __forceinline__ void gemm_core(const float* __restrict__ A,
                                          const float* __restrict__ B, int K,
                                          int N, int mwave, int nblk,
                                          int nwave_off, float* BtLds,
                                          v8f acc[4]) {
  const int tid = threadIdx.x;
  const int lane = tid & 31;
  const int half = lane >> 4;
  const int l16 = lane & 15;
  const float* Arow = A + (size_t)(mwave + l16) * K;

  acc[0] = (v8f){};
  acc[1] = (v8f){};
  acc[2] = (v8f){};
  acc[3] = (v8f){};

  issue_btile(B, N, 0, nblk, BtLds, tid);
  wait_async0();
  __syncthreads();

  const int nk = K / KC;
  for (int t = 0; t < nk; ++t) {
    const int kbase = t * KC;
    const float* cur = BtLds + (t & 1) * (BN * KC);
    if (t + 1 < nk)
      issue_btile(B, N, kbase + KC, nblk, BtLds + ((t + 1) & 1) * (BN * KC),
                  tid);

    const float* brow = cur + (nwave_off + l16) * KC;  // Bt[n][*]
#pragma unroll
    for (int k = 0; k < KC; k += 4) {
      const int ka = k + (half << 1);
      v2f b = *(const v2f*)(brow + ka);  // ds_load_b64
      const size_t kaA = (size_t)(kbase + ka);
      v2f a0 = *(const v2f*)(Arow + kaA);
      v2f a1 = *(const v2f*)(Arow + (size_t)16 * K + kaA);
      v2f a2 = *(const v2f*)(Arow + (size_t)32 * K + kaA);
      v2f a3 = *(const v2f*)(Arow + (size_t)48 * K + kaA);
      acc[0] = wmma_f32_4(a0, b, acc[0]);
      acc[1] = wmma_f32_4(a1, b, acc[1]);
      acc[2] = wmma_f32_4(a2, b, acc[2]);
      acc[3] = wmma_f32_4(a3, b, acc[3]);
    }
    wait_async0();
    __syncthreads();
  }
}

// ---------------------------------------------------------------------------
// Square-ish GEMM: C = alpha * A @ B ; optionally Eacc += C.
__global__ __launch_bounds__(NTHREADS) void gemm_f32_wmma_kernel(
    const float* __restrict__ A, const float* __restrict__ B,
    float* __restrict__ C, float* __restrict__ Eacc, int K, int N,
    float alpha) {
  __shared__ float Bt[2 * BN * KC];
  const int tid = threadIdx.x;
  const int wave = tid >> 5;
  const int wr = wave >> 1;  // wave row (0..1)
  const int wc = wave & 1;   // wave col (0..1)
  const int lane = tid & 31;
  const int half = lane >> 4;
  const int l16 = lane & 15;

  const int m0 = blockIdx.x * BM + wr * 64;
  const int nblk = blockIdx.y * BN;
  const int nw = nblk + wc * 16;

  v8f acc[4];
  gemm_core(A, B, K, N, m0, nblk, wc * 16, Bt, acc);

#pragma unroll
  for (int i = 0; i < 4; ++i) {
    const size_t base = (size_t)(m0 + (i << 4) + (half << 3)) * N + nw + l16;
#pragma unroll
    for (int r = 0; r < 8; ++r) {
      const float v = acc[i][r] * alpha;
      const size_t idx = base + (size_t)r * N;
      C[idx] = v;
      if (Eacc) Eacc[idx] += v;
    }
  }
}

// Main transform: out = (rnorm[m] * (X @ U) + bias[n])^2 * gate[n]
__global__ __launch_bounds__(NTHREADS) void xform_gemm_kernel(
    const float* __restrict__ X, const float* __restrict__ U,
    const float* __restrict__ rnorm, const float* __restrict__ bias,
    const float* __restrict__ gate, float* __restrict__ out) {
  __shared__ float Bt[2 * BN * KC];
  const int tid = threadIdx.x;
  const int wave = tid >> 5;
  const int wr = wave >> 1;
  const int wc = wave & 1;
  const int lane = tid & 31;
  const int half = lane >> 4;
  const int l16 = lane & 15;

  const int m0 = blockIdx.x * BM + wr * 64;
  const int nblk = blockIdx.y * BN;
  const int nw = nblk + wc * 16;

  v8f acc[4];
  gemm_core(X, U, DQ, DQ, m0, nblk, wc * 16, Bt, acc);

  const float bb = bias[nw + l16];
  const float gg = gate[nw + l16];

#pragma unroll
  for (int i = 0; i < 4; ++i) {
    const int mrow = m0 + (i << 4) + (half << 3);
    const size_t base = (size_t)mrow * DQ + nw + l16;
#pragma unroll
    for (int r = 0; r < 8; ++r) {
      const float t = acc[i][r] * rnorm[mrow + r] + bb;
      out[base + (size_t)r * DQ] = t * t * gg;
    }
  }
}

// ---------------------------------------------------------------------------
// A[i][j] = scale * (M[i][j] - M[j][i])
__global__ void skew_scale_kernel(const float* __restrict__ M,
                                  float* __restrict__ A, float scale) {
  int idx = blockIdx.x * blockDim.x + threadIdx.x;
  int row = idx >> 10;
  int col = idx & (DQ - 1);
  A[idx] = scale * (M[idx] - M[col * DQ + row]);
}

// E = I + As
__global__ void expm_init_kernel(const float* __restrict__ As,
                                 float* __restrict__ E) {
  int idx = blockIdx.x * blockDim.x + threadIdx.x;
  int row = idx >> 10;
  int col = idx & (DQ - 1);
  E[idx] = As[idx] + ((row == col) ? 1.0f : 0.0f);
}

// rnorm[row] = rsqrt(max(sum(x^2), 1e-12))
__global__ void rownorm_kernel(const float* __restrict__ X,
                               float* __restrict__ rnorm) {
  __shared__ float red[256];
  int row = blockIdx.x;
  const float* x = X + (size_t)row * DQ;
  float s = 0.0f;
  for (int j = threadIdx.x; j < DQ; j += 256) {
    float v = x[j];
    s += v * v;
  }
  red[threadIdx.x] = s;
  __syncthreads();
  for (int off = 128; off > 0; off >>= 1) {
    if (threadIdx.x < off) red[threadIdx.x] += red[threadIdx.x + off];
    __syncthreads();
  }
  if (threadIdx.x == 0) rnorm[row] = rsqrtf(fmaxf(red[0], 1e-12f));
}

// gate = softmax(g) over 1024; also written to the output tail.
__global__ void softmax_kernel(const float* __restrict__ g,
                               float* __restrict__ gate_ws,
                               float* __restrict__ gate_out) {
  __shared__ float red[1024];
  int t = threadIdx.x;
  float v = g[t];
  red[t] = v;
  __syncthreads();
  for (int off = 512; off > 0; off >>= 1) {
    if (t < off) red[t] = fmaxf(red[t], red[t + off]);
    __syncthreads();
  }
  float mx = red[0];
  __syncthreads();
  float e = expf(v - mx);
  red[t] = e;
  __syncthreads();
  for (int off = 512; off > 0; off >>= 1) {
    if (t < off) red[t] += red[t + off];
    __syncthreads();
  }
  float p = e / red[0];
  gate_ws[t] = p;
  gate_out[t] = p;
}

// ---------------------------------------------------------------------------
extern "C" void kernel_launch(void* const* d_in, const int* in_sizes, int n_in,
                              void* d_out, int out_size, void* d_ws,
                              size_t ws_size, hipStream_t stream) {
  (void)in_sizes;
  (void)n_in;
  (void)out_size;
  (void)ws_size;

  const float* X = (const float*)d_in[0];       // [8,4096,1024]
  const float* M_real = (const float*)d_in[1];  // [1024,1024]
  const float* M_imag = (const float*)d_in[2];  // [1024,1024]
  const float* bias = (const float*)d_in[3];    // [1024]
  const float* qgate = (const float*)d_in[4];   // [1024]
  float* out = (float*)d_out;                   // gated_output then gate_probs
  float* out_gate = out + (size_t)NROWS * DQ;

  float* w = (float*)d_ws;
  float* As = w;            // scaled skew(M_real)
  float* Bs = w + DDQ;      // scaled skew(M_imag)
  float* E0 = w + 2 * DDQ;  // expm(A) accumulator / result
  float* E1 = w + 3 * DDQ;  // squaring ping-pong
  float* P0 = w + 4 * DDQ;  // Taylor term ping-pong
  float* P1 = w + 5 * DDQ;
  float* EB = w + 6 * DDQ;      // expm(B) accumulator / result
  float* Ub = w + 7 * DDQ;      // U = expm(A) @ expm(B)
  float* rnorm = w + 8 * DDQ;   // [32768]
  float* gate = rnorm + NROWS;  // [1024]

  const int ew = (int)(DDQ / 256);
  const dim3 sq_grid(DQ / BM, DQ / BN);

  auto gemm = [&](const float* A_, const float* B_, float* C_, float* E_,
                  float alpha) {
    gemm_f32_wmma_kernel<<<sq_grid, NTHREADS, 0, stream>>>(A_, B_, C_, E_, DQ,
                                                           DQ, alpha);
  };

  skew_scale_kernel<<<ew, 256, 0, stream>>>(M_real, As, EXPM_SCALE);
  skew_scale_kernel<<<ew, 256, 0, stream>>>(M_imag, Bs, EXPM_SCALE);

  // --- expm(As) -> E0 ---
  expm_init_kernel<<<ew, 256, 0, stream>>>(As, E0);
  {
    const float* P = As;
    float* T = P0;
    float* Tn = P1;
    for (int k = 2; k <= TAYLOR_ORDER; ++k) {
      gemm(P, As, T, E0, 1.0f / (float)k);  // T = (1/k) P@As ; E0 += T
      P = T;
      float* tmp = T;
      T = Tn;
      Tn = tmp;
    }
    float* cur = E0;
    float* oth = E1;
    for (int i = 0; i < SQUARINGS; ++i) {
      gemm(cur, cur, oth, nullptr, 1.0f);
      float* tmp = cur;
      cur = oth;
      oth = tmp;
    }
    // SQUARINGS even -> result back in E0
  }

  // --- expm(Bs) -> EB ---
  expm_init_kernel<<<ew, 256, 0, stream>>>(Bs, EB);
  {
    const float* P = Bs;
    float* T = P0;
    float* Tn = P1;
    for (int k = 2; k <= TAYLOR_ORDER; ++k) {
      gemm(P, Bs, T, EB, 1.0f / (float)k);
      P = T;
      float* tmp = T;
      T = Tn;
      Tn = tmp;
    }
    float* cur = EB;
    float* oth = E1;
    for (int i = 0; i < SQUARINGS; ++i) {
      gemm(cur, cur, oth, nullptr, 1.0f);
      float* tmp = cur;
      cur = oth;
      oth = tmp;
    }
    // result back in EB
  }

  // --- U = expm(A) @ expm(B) ---
  gemm(E0, EB, Ub, nullptr, 1.0f);

  // --- epilogue inputs ---
  rownorm_kernel<<<NROWS, 256, 0, stream>>>(X, rnorm);
  softmax_kernel<<<1, 1024, 0, stream>>>(qgate, gate, out_gate);

  // --- main transform GEMM ---
  const dim3 xg(NROWS / BM, DQ / BN);
  xform_gemm_kernel<<<xg, NTHREADS, 0, stream>>>(X, Ub, rnorm, bias, gate, out);
}